// StabilizerEmbedder_29643864277020
// MI455X (gfx1250) — compile-verified
//
#include <hip/hip_runtime.h>
#include <hip/hip_bf16.h>

typedef float v2f __attribute__((ext_vector_type(2)));
typedef float v8f __attribute__((ext_vector_type(8)));

#define HDIM 256
#define NFEAT 8

// One wave = 16-row x 256-col tile, single pass over the output.
// GEMM with augmented K=12:
//   A = [ x[r,0:8] | 1 | f1 | f2 | 0 ]          (f1 = mask==1, f2 = mask==2)
//   B = [ W[0:8,:] ; bsum ; final_on ; final_off ; 0 ]
// so D already contains x*W + bsum + f1*final_on + f2*final_off.
// Epilogue: + index_embed[s,:], LayerNorm via 16-lane __shfl_xor reduction,
// then out = (h-mu)*rstd*gamma + beta in 2 FMAs/element.
__global__ __launch_bounds__(256) void stab_embed_kernel(
    const float* __restrict__ x,          // [B*S, 8]
    const long long* __restrict__ mask,   // [B*S]
    const float* __restrict__ W,          // [8, 256]
    const float* __restrict__ bias,       // [8, 256]
    const float* __restrict__ embed,      // [S, 256]
    const float* __restrict__ fon,        // [256]
    const float* __restrict__ foff,       // [256]
    const float* __restrict__ gmm,        // [256]
    const float* __restrict__ bet,        // [256]
    float* __restrict__ out,              // [B*S, 256]
    int S)
{
    // s_bc0[n] = sum_f bias[f][n], s_bc0[256+n] = final_off[n]   (B aug row K=8 / K=10)
    // s_bc1[n] = final_on[n],      s_bc1[256+n] = 0              (B aug row K=9 / K=11)
    __shared__ float s_bc0[2*HDIM], s_bc1[2*HDIM], s_g[HDIM], s_be[HDIM];
    const int tid = threadIdx.x;
    {
        float bs = 0.f;
        #pragma unroll
        for (int f = 0; f < NFEAT; ++f) bs += bias[f*HDIM + tid];
        s_bc0[tid]        = bs;
        s_bc0[HDIM + tid] = foff[tid];
        s_bc1[tid]        = fon[tid];
        s_bc1[HDIM + tid] = 0.f;
        s_g[tid]  = gmm[tid];
        s_be[tid] = bet[tid];
    }
    __syncthreads();

    const int lane = tid & 31;
    const int wv   = tid >> 5;
    const int hlf  = lane >> 4;
    const int mrow = lane & 15;

    const long long r0 = ((long long)blockIdx.x * 8 + wv) * 16;
    const int s0 = (int)(r0 % S);

    // Prefetch this tile's 16KB of index_embed (L2-hot) while the GEMM runs.
    {
        const char* p = (const char*)(embed + (long long)s0 * HDIM);
        #pragma unroll
        for (int i = 0; i < 4; ++i)
            __builtin_prefetch(p + (i * 32 + lane) * 128, 0, 3);
    }

    // ---- A operands (ISA 16x4 f32 A layout: half0 -> K pair {0,1}, half1 -> {2,3}) ----
    const float* xr = x + (r0 + mrow) * NFEAT + hlf * 2;
    v2f a0, a1, a2;
    a0[0] = xr[0]; a0[1] = xr[1];     // K 0..3
    a1[0] = xr[4]; a1[1] = xr[5];     // K 4..7
    {                                  // K 8..11 = [1, f1, f2, 0] for row mrow
        const long long mv = mask[r0 + mrow];
        const float f1 = (mv == 1) ? 1.f : 0.f;
        const float f2 = (mv == 2) ? 1.f : 0.f;
        a2[0] = hlf ? f2  : 1.f;
        a2[1] = hlf ? 0.f : f1;
    }

    // ---- GEMM: 16 N-tiles, 3 K-chunks each ----
    const int kb = hlf * 2;
    const float* pb2x = s_bc0 + hlf * HDIM;   // half0: bsum, half1: final_off
    const float* pb2y = s_bc1 + hlf * HDIM;   // half0: final_on, half1: zeros
    v8f acc[16];
    #pragma unroll
    for (int t = 0; t < 16; ++t) {
        const int n = t * 16 + mrow;
        v2f b0, b1, b2;
        b0[0] = W[(kb    ) * HDIM + n];
        b0[1] = W[(kb + 1) * HDIM + n];
        b1[0] = W[(kb + 4) * HDIM + n];
        b1[1] = W[(kb + 5) * HDIM + n];
        b2[0] = pb2x[n];
        b2[1] = pb2y[n];
        v8f c = {0.f, 0.f, 0.f, 0.f, 0.f, 0.f, 0.f, 0.f};
        c = __builtin_amdgcn_wmma_f32_16x16x4_f32(false, a0, false, b0, (short)0, c, false, false);
        c = __builtin_amdgcn_wmma_f32_16x16x4_f32(false, a1, false, b1, (short)0, c, false, false);
        c = __builtin_amdgcn_wmma_f32_16x16x4_f32(false, a2, false, b2, (short)0, c, false, false);
        acc[t] = c;
    }

    // ---- Phase 1: += index_embed, accumulate sum / sum-of-squares per row ----
    // D layout: VGPR v of lane holds row (v + 8*hlf), column (t*16 + mrow).
    const float* emb = embed + (long long)(s0 + hlf * 8) * HDIM + mrow;
    float sum[8] = {0.f, 0.f, 0.f, 0.f, 0.f, 0.f, 0.f, 0.f};
    float ssq[8] = {0.f, 0.f, 0.f, 0.f, 0.f, 0.f, 0.f, 0.f};
    #pragma unroll
    for (int t = 0; t < 16; ++t) {
        #pragma unroll
        for (int v = 0; v < 8; ++v) {
            float val = acc[t][v] + emb[v * HDIM + t * 16];
            acc[t][v] = val;
            sum[v] += val;
            ssq[v] = fmaf(val, val, ssq[v]);
        }
    }

    // ---- Row statistics: reduce across the 16 lanes of each half-wave ----
    float scl[8], off[8];
    #pragma unroll
    for (int v = 0; v < 8; ++v) {
        float s_ = sum[v], q = ssq[v];
        #pragma unroll
        for (int o = 1; o < 16; o <<= 1) {
            s_ += __shfl_xor(s_, o, 32);
            q  += __shfl_xor(q,  o, 32);
        }
        const float mean = s_ * (1.f / 256.f);
        const float var  = fmaf(-mean, mean, q * (1.f / 256.f));  // ddof=0
        const float rstd = rsqrtf(var + 1e-5f);
        scl[v] = rstd;
        off[v] = -mean * rstd;
    }

    // ---- Phase 2: normalize, affine, store (2 FMA + 1 store per element) ----
    float* orow = out + (r0 + hlf * 8) * HDIM + mrow;
    #pragma unroll
    for (int t = 0; t < 16; ++t) {
        const int n = t * 16 + mrow;
        const float g  = s_g[n];
        const float be = s_be[n];
        #pragma unroll
        for (int v = 0; v < 8; ++v) {
            const float val = fmaf(acc[t][v], scl[v], off[v]);
            orow[v * HDIM + t * 16] = fmaf(val, g, be);
        }
    }
}

extern "C" void kernel_launch(void* const* d_in, const int* in_sizes, int n_in,
                              void* d_out, int out_size, void* d_ws, size_t ws_size,
                              hipStream_t stream) {
    (void)n_in; (void)out_size; (void)d_ws; (void)ws_size;
    const float*     x     = (const float*)d_in[0];
    const long long* fmask = (const long long*)d_in[1];   // int64
    const float*     W     = (const float*)d_in[2];
    const float*     bias  = (const float*)d_in[3];
    const float*     embed = (const float*)d_in[4];
    const float*     fon   = (const float*)d_in[5];
    const float*     foff  = (const float*)d_in[6];
    const float*     gmm   = (const float*)d_in[7];
    const float*     bet   = (const float*)d_in[8];
    float*           out   = (float*)d_out;

    const int R = in_sizes[1];            // B*S total rows
    const int S = in_sizes[4] / HDIM;     // index_embed rows
    const int blocks = R / 128;           // 8 waves/block * 16 rows/wave

    stab_embed_kernel<<<blocks, 256, 0, stream>>>(
        x, fmask, W, bias, embed, fon, foff, gmm, bet, out, S);
}